// GCNModel_16733192585316
// MI455X (gfx1250) — compile-verified
//
#include <hip/hip_runtime.h>

typedef float v2f __attribute__((ext_vector_type(2)));
typedef float v8f __attribute__((ext_vector_type(8)));

// D = A(16x4, f32) * B(4x16, f32) + C(16x16, f32) on the CDNA5 matrix pipe.
// args: (neg_a, A, neg_b, B, c_mod, C, reuse_a, reuse_b)
#define WMMA_F32X4(a, b, c) \
  __builtin_amdgcn_wmma_f32_16x16x4_f32(false, (a), false, (b), (short)0, (c), false, false)

__global__ void k_fill(float* __restrict__ p, float v, int n) {
  int i = blockIdx.x * blockDim.x + threadIdx.x;
  if (i < n) p[i] = v;
}

__global__ void k_deg_count(const long long* __restrict__ dst, float* __restrict__ deg, int E) {
  int e = blockIdx.x * blockDim.x + threadIdx.x;
  if (e < E) atomicAdd(deg + (int)dst[e], 1.0f);
}

__global__ void k_rsqrt(const float* __restrict__ deg, float* __restrict__ dinv, int n) {
  int i = blockIdx.x * blockDim.x + threadIdx.x;
  if (i < n) dinv[i] = rsqrtf(deg[i]);
}

// lin1[n, j] = x[n,0]*W1[0,j] + x[n,1]*W1[1,j]   (K=2: scalar FMA is optimal)
__global__ void k_lin1(const float* __restrict__ x, const float* __restrict__ W1,
                       float* __restrict__ lin1, int N) {
  int i = blockIdx.x * blockDim.x + threadIdx.x;
  if (i >= N * 64) return;
  int n = i >> 6, j = i & 63;
  lin1[i] = fmaf(x[n * 2], W1[j], x[n * 2 + 1] * W1[64 + j]);
}

// self-loop term: agg[n,j] = lin[n,j] / deg[n]
__global__ void k_self_init(const float* __restrict__ lin, const float* __restrict__ deg,
                            float* __restrict__ agg, int N, int logF) {
  int i = blockIdx.x * blockDim.x + threadIdx.x;
  if (i >= (N << logF)) return;
  agg[i] = lin[i] / deg[i >> logF];
}

// per edge, per 4-feature chunk (blockIdx.y): agg[dst] += lin[src] * dinv[src]*dinv[dst]
__global__ void k_scatter(const long long* __restrict__ src, const long long* __restrict__ dstv,
                          const float* __restrict__ dinv, const float* __restrict__ lin,
                          float* __restrict__ agg, int E, int F) {
  int e = blockIdx.x * blockDim.x + threadIdx.x;
  if (e >= E) return;
  int q4 = blockIdx.y << 2;
  int s = (int)src[e], d = (int)dstv[e];
  float norm = dinv[s] * dinv[d];
  const float4 v = *(const float4*)(lin + s * F + q4);
  float* b = agg + d * F + q4;
  atomicAdd(b + 0, v.x * norm);
  atomicAdd(b + 1, v.y * norm);
  atomicAdd(b + 2, v.z * norm);
  atomicAdd(b + 3, v.w * norm);
}

__global__ void k_bias_relu(float* __restrict__ h, const float* __restrict__ bias, int N, int logF) {
  int i = blockIdx.x * blockDim.x + threadIdx.x;
  if (i >= (N << logF)) return;
  h[i] = fmaxf(h[i] + bias[i & ((1 << logF) - 1)], 0.0f);
}

// segment max: values are post-ReLU (>=0) so int-compare atomicMax is order-correct
__global__ void k_pool(const float* __restrict__ h2, const long long* __restrict__ batch,
                       float* __restrict__ gpool, int N) {
  int i = blockIdx.x * blockDim.x + threadIdx.x;
  if (i >= N * 32) return;
  int n = i >> 5, j = i & 31;
  int g = (int)batch[n];
  atomicMax((int*)(gpool + g * 32 + j), __float_as_int(h2[i]));
}

// One wave32 computes a 16-row x NCOLS (<=32) tile of C = A[M,K] @ B[K,NCOLS]
// with V_WMMA_F32_16X16X4_F32. K/NCOLS compile-time -> fully unrolled WMMA chain.
// B is staged in LDS once per block and read as fragments via ds_load.
// A fragment: lane -> (row = lane&15, K-pair = (lane>>4)*2). B symmetric on columns.
// C fragment: VGPR v -> row (lane>>4)*8 + v, col lane&15.
template <int K, int NCOLS, bool BIAS, bool RELU>
__global__ void k_gemm_t(const float* __restrict__ A, int lda,
                         const float* __restrict__ B,
                         const float* __restrict__ bias,
                         float* __restrict__ C, int ldc, int M) {
  __shared__ float Bs[K * NCOLS];
  for (int i = threadIdx.x; i < K * NCOLS; i += blockDim.x) Bs[i] = B[i];
  __syncthreads();

  int wave = threadIdx.x >> 5;
  int lane = threadIdx.x & 31;
  int tile = blockIdx.x * (blockDim.x >> 5) + wave;
  int m0 = tile << 4;
  if (m0 >= M) return;  // wave-uniform

  int row = lane & 15;           // A row within tile / B-C column within half-tile
  int khalf = (lane >> 4) << 1;  // 0 or 2
  int ar = m0 + row;
  if (ar >= M) ar = M - 1;       // safe A read for ragged M
  const float* __restrict__ Arow = A + (size_t)ar * lda;

  v8f c0 = {}, c1 = {};
#pragma unroll
  for (int k0 = 0; k0 < K; k0 += 4) {
    int kk = k0 + khalf;
    v2f a;
    a.x = Arow[kk];
    a.y = Arow[kk + 1];
    v2f b0, b1;
    b0.x = Bs[kk * NCOLS + row];
    b0.y = Bs[(kk + 1) * NCOLS + row];
    int col2 = row + 16;
    if (NCOLS >= 32 || col2 < NCOLS) {
      b1.x = Bs[kk * NCOLS + col2];
      b1.y = Bs[(kk + 1) * NCOLS + col2];
    } else {
      b1.x = 0.0f;
      b1.y = 0.0f;
    }
    c0 = WMMA_F32X4(a, b0, c0);
    c1 = WMMA_F32X4(a, b1, c1);
  }

  int mh = (lane >> 4) << 3;
#pragma unroll
  for (int v = 0; v < 8; ++v) {
    int m = m0 + mh + v;
    if (m >= M) continue;
    float x0 = c0[v];
    if (BIAS) x0 += bias[row];
    if (RELU) x0 = fmaxf(x0, 0.0f);
    C[m * ldc + row] = x0;
    int col2 = row + 16;
    if (col2 < NCOLS) {
      float x1 = c1[v];
      if (BIAS) x1 += bias[col2];
      if (RELU) x1 = fmaxf(x1, 0.0f);
      C[m * ldc + col2] = x1;
    }
  }
}

static inline int cdiv(int a, int b) { return (a + b - 1) / b; }

extern "C" void kernel_launch(void* const* d_in, const int* in_sizes, int n_in,
                              void* d_out, int out_size, void* d_ws, size_t ws_size,
                              hipStream_t stream) {
  const float*     x     = (const float*)d_in[0];
  const long long* ei    = (const long long*)d_in[1];
  const long long* batch = (const long long*)d_in[2];
  const float*     W1    = (const float*)d_in[3];
  const float*     b1    = (const float*)d_in[4];
  const float*     W2    = (const float*)d_in[5];
  const float*     b2    = (const float*)d_in[6];
  const float*     W3    = (const float*)d_in[7];
  const float*     b3    = (const float*)d_in[8];
  const float*     W4    = (const float*)d_in[9];
  const float*     b4    = (const float*)d_in[10];
  float* out = (float*)d_out;

  int N = in_sizes[0] / 2;   // 100000
  int E = in_sizes[1] / 2;   // 3200000
  int G = out_size / 28;     // 256

  const long long* src = ei;
  const long long* dst = ei + E;

  // ws layout (floats): deg[N] | dinv[N] | bufL[64N] | bufH[64N] | gpool[G*32] | g3[G*32]
  float* ws    = (float*)d_ws;
  float* deg   = ws;
  float* dinv  = ws + (size_t)N;
  float* bufL  = ws + 2 * (size_t)N;
  float* bufH  = bufL + 64 * (size_t)N;
  float* gpool = bufH + 64 * (size_t)N;
  float* g3    = gpool + (size_t)G * 32;

  float* lin1 = bufL;                    // [N,64]
  float* agg1 = bufH;                    // [N,64] -> h1
  float* lin2 = bufL;                    // [N,32] (lin1 dead by then)
  float* agg2 = bufL + 32 * (size_t)N;   // [N,32] -> h2

  const int T = 256;

  // degrees (with self loop) and D^{-1/2}
  k_fill<<<cdiv(N, T), T, 0, stream>>>(deg, 1.0f, N);
  k_deg_count<<<cdiv(E, T), T, 0, stream>>>(dst, deg, E);
  k_rsqrt<<<cdiv(N, T), T, 0, stream>>>(deg, dinv, N);

  // layer 1 (64 features)
  k_lin1<<<cdiv(N * 64, T), T, 0, stream>>>(x, W1, lin1, N);
  k_self_init<<<cdiv(N * 64, T), T, 0, stream>>>(lin1, deg, agg1, N, 6);
  {
    dim3 g(cdiv(E, T), 16);
    k_scatter<<<g, T, 0, stream>>>(src, dst, dinv, lin1, agg1, E, 64);
  }
  k_bias_relu<<<cdiv(N * 64, T), T, 0, stream>>>(agg1, b1, N, 6);

  // layer 2 linear: lin2 = h1 @ W2  (WMMA f32 16x16x4; 6250 exact M-tiles; 8 waves/block)
  {
    int tiles = cdiv(N, 16);
    k_gemm_t<64, 32, false, false>
        <<<cdiv(tiles, 8), 256, 0, stream>>>(agg1, 64, W2, nullptr, lin2, 32, N);
  }
  k_self_init<<<cdiv(N * 32, T), T, 0, stream>>>(lin2, deg, agg2, N, 5);
  {
    dim3 g(cdiv(E, T), 8);
    k_scatter<<<g, T, 0, stream>>>(src, dst, dinv, lin2, agg2, E, 32);
  }
  k_bias_relu<<<cdiv(N * 32, T), T, 0, stream>>>(agg2, b2, N, 5);

  // global max pool -> [G,32]
  k_fill<<<cdiv(G * 32, T), T, 0, stream>>>(gpool, 0.0f, G * 32);
  k_pool<<<cdiv(N * 32, T), T, 0, stream>>>(agg2, batch, gpool, N);

  // g3 = relu(gpool @ W3 + b3); out = relu(g3 @ W4 + b4)  (WMMA, 16 M-tiles each)
  k_gemm_t<32, 32, true, true>
      <<<cdiv(cdiv(G, 16), 8), 256, 0, stream>>>(gpool, 32, W3, b3, g3, 32, G);
  k_gemm_t<32, 28, true, true>
      <<<cdiv(cdiv(G, 16), 8), 256, 0, stream>>>(g3, 32, W4, b4, out, 28, G);
}